// GCN_23029614641915
// MI455X (gfx1250) — compile-verified
//
#include <hip/hip_runtime.h>

#define NUM_USERS 50000
#define NUM_ITEMS 50000
#define N_NODES   100000
#define DIM_FEAT  256
#define HID       128
#define EMB       128
#define NLAYERS   3
#define N_EDGES   1600000

typedef __attribute__((ext_vector_type(2))) float v2f;
typedef __attribute__((ext_vector_type(8))) float v8f;

__device__ __forceinline__ float lrelu(float x) { return x > 0.0f ? x : 0.01f * x; }

// ---------------------------------------------------------------------------
// Copy user embeddings into X rows [0, NUM_USERS) and into the second output.
// ---------------------------------------------------------------------------
__global__ void gcn_copy_users(const float* __restrict__ ue,
                               float* __restrict__ X,
                               float* __restrict__ out2) {
    int i = blockIdx.x * blockDim.x + threadIdx.x;          // float4 index
    if (i >= NUM_USERS * HID / 4) return;
    float4 v = ((const float4*)ue)[i];
    ((float4*)X)[i]    = v;
    ((float4*)out2)[i] = v;
}

// ---------------------------------------------------------------------------
// Item transform: X[NUM_USERS+r, :] = feat[r,:] @ trans_w + trans_b
// 16-row tile per block, 8 waves, one 16-col WMMA tile per wave, K=256.
// ---------------------------------------------------------------------------
__global__ __launch_bounds__(256) void gcn_trans_items(
        const float* __restrict__ feat, const float* __restrict__ W,
        const float* __restrict__ bias, float* __restrict__ X) {
    __shared__ float featT[16][DIM_FEAT + 1];
    const int tid = threadIdx.x;
    const int r0  = blockIdx.x * 16;
    for (int i = tid; i < 16 * DIM_FEAT; i += 256) {
        int r = i >> 8, c = i & (DIM_FEAT - 1);
        featT[r][c] = feat[(long long)(r0 + r) * DIM_FEAT + c];
    }
    __syncthreads();

    const int lane = tid & 31, wave = tid >> 5;
    const int hf = lane >> 4, mn = lane & 15;
    const float* Bw = W + 16 * wave;                        // column offset
    v8f acc = {};
    for (int k = 0; k < DIM_FEAT; k += 4) {
        v2f a, b;
        a.x = featT[mn][k + 2 * hf + 0];
        a.y = featT[mn][k + 2 * hf + 1];
        b.x = Bw[(k + 2 * hf + 0) * HID + mn];
        b.y = Bw[(k + 2 * hf + 1) * HID + mn];
        acc = __builtin_amdgcn_wmma_f32_16x16x4_f32(false, a, false, b,
                                                    (short)0, acc, false, false);
    }
    const int col = 16 * wave + mn;
    const float bb = bias[col];
#pragma unroll
    for (int v = 0; v < 8; ++v) {
        int row = v + 8 * hf;
        X[(long long)(NUM_USERS + r0 + row) * HID + col] = acc[v] + bb;
    }
}

// ---------------------------------------------------------------------------
// Row-wise L2 normalize: one wave per 128-float row.
// ---------------------------------------------------------------------------
__global__ void gcn_l2norm(float* __restrict__ X) {
    const int wave = threadIdx.x >> 5, lane = threadIdx.x & 31;
    const long long row = (long long)blockIdx.x * 8 + wave;
    float4* p = (float4*)(X + row * HID) + lane;
    float4 v = *p;
    float ss = v.x * v.x + v.y * v.y + v.z * v.z + v.w * v.w;
#pragma unroll
    for (int off = 16; off; off >>= 1) ss += __shfl_xor(ss, off, 32);
    float inv = 1.0f / fmaxf(sqrtf(ss), 1e-12f);
    v.x *= inv; v.y *= inv; v.z *= inv; v.w *= inv;
    *p = v;
}

// ---------------------------------------------------------------------------
// Zero AGG.
// ---------------------------------------------------------------------------
__global__ void gcn_zero(float4* __restrict__ p, int n4) {
    int i = blockIdx.x * blockDim.x + threadIdx.x;
    if (i < n4) p[i] = make_float4(0.f, 0.f, 0.f, 0.f);
}

// ---------------------------------------------------------------------------
// Edge scatter-add: one wave per edge, coalesced 128B gather, 4 f32 atomics
// per lane into L2-resident AGG.
// ---------------------------------------------------------------------------
__global__ void gcn_edge_scatter(const int* __restrict__ src,
                                 const int* __restrict__ dst,
                                 const float* __restrict__ X,
                                 float* __restrict__ AGG) {
    long long gid = (long long)blockIdx.x * blockDim.x + threadIdx.x;
    int e = (int)(gid >> 5);
    int lane = (int)(gid & 31);
    if (e >= N_EDGES) return;
    const float* xs = X   + (long long)src[e] * HID;
    float*       ad = AGG + (long long)dst[e] * HID;
#pragma unroll
    for (int j = 0; j < 4; ++j)
        unsafeAtomicAdd(&ad[lane + 32 * j], xs[lane + 32 * j]);
}

// ---------------------------------------------------------------------------
// Fused layer: 16-row tile per block, 8 waves (one 16-col tile each).
//   h = lrelu(agg @ Wh)                        (K=128)
//   u = lrelu(h @ Wd + bd + node_emb)          (K=128)
//   x = lrelu([h|u] @ Wo + bo)                 (K=256, split over h/u halves)
// h and u tiles live in LDS only; weights streamed from L2.
// ---------------------------------------------------------------------------
__global__ __launch_bounds__(256) void gcn_layer(
        const float* __restrict__ AGG,
        const float* __restrict__ Wh,
        const float* __restrict__ Wd, const float* __restrict__ bd,
        const float* __restrict__ Wo, const float* __restrict__ bo,
        const float* __restrict__ node_emb,
        float* __restrict__ Xout) {
    __shared__ float aggT[16][HID + 1];
    __shared__ float hT[16][HID + 1];
    __shared__ float uT[16][HID + 1];

    const int tid = threadIdx.x;
    const int r0  = blockIdx.x * 16;
    for (int i = tid; i < 16 * HID; i += 256) {
        int r = i >> 7, c = i & (HID - 1);
        aggT[r][c] = AGG[(long long)(r0 + r) * HID + c];
    }
    __syncthreads();

    const int lane = tid & 31, wave = tid >> 5;
    const int hf = lane >> 4, mn = lane & 15;
    const int col = 16 * wave + mn;

    // ---- h = lrelu(agg @ Wh) ----
    {
        v8f acc = {};
        const float* B = Wh + 16 * wave;
        for (int k = 0; k < HID; k += 4) {
            v2f a, b;
            a.x = aggT[mn][k + 2 * hf + 0];
            a.y = aggT[mn][k + 2 * hf + 1];
            b.x = B[(k + 2 * hf + 0) * HID + mn];
            b.y = B[(k + 2 * hf + 1) * HID + mn];
            acc = __builtin_amdgcn_wmma_f32_16x16x4_f32(false, a, false, b,
                                                        (short)0, acc, false, false);
        }
#pragma unroll
        for (int v = 0; v < 8; ++v) hT[v + 8 * hf][col] = lrelu(acc[v]);
    }
    __syncthreads();

    // ---- u = lrelu(h @ Wd + bd + node_emb) ----
    {
        v8f acc = {};
        const float* B = Wd + 16 * wave;
        for (int k = 0; k < HID; k += 4) {
            v2f a, b;
            a.x = hT[mn][k + 2 * hf + 0];
            a.y = hT[mn][k + 2 * hf + 1];
            b.x = B[(k + 2 * hf + 0) * HID + mn];
            b.y = B[(k + 2 * hf + 1) * HID + mn];
            acc = __builtin_amdgcn_wmma_f32_16x16x4_f32(false, a, false, b,
                                                        (short)0, acc, false, false);
        }
        const float bb = bd[col];
#pragma unroll
        for (int v = 0; v < 8; ++v) {
            int row = v + 8 * hf;
            float ne = node_emb[(long long)(r0 + row) * EMB + col];
            uT[row][col] = lrelu(acc[v] + bb + ne);
        }
    }
    __syncthreads();

    // ---- x = lrelu([h|u] @ Wo + bo) ----
    {
        v8f acc = {};
        const float* B0 = Wo + 16 * wave;               // rows 0..127   (h half)
        const float* B1 = Wo + HID * EMB + 16 * wave;   // rows 128..255 (u half)
        for (int k = 0; k < HID; k += 4) {
            v2f a, b;
            a.x = hT[mn][k + 2 * hf + 0];
            a.y = hT[mn][k + 2 * hf + 1];
            b.x = B0[(k + 2 * hf + 0) * EMB + mn];
            b.y = B0[(k + 2 * hf + 1) * EMB + mn];
            acc = __builtin_amdgcn_wmma_f32_16x16x4_f32(false, a, false, b,
                                                        (short)0, acc, false, false);
        }
        for (int k = 0; k < HID; k += 4) {
            v2f a, b;
            a.x = uT[mn][k + 2 * hf + 0];
            a.y = uT[mn][k + 2 * hf + 1];
            b.x = B1[(k + 2 * hf + 0) * EMB + mn];
            b.y = B1[(k + 2 * hf + 1) * EMB + mn];
            acc = __builtin_amdgcn_wmma_f32_16x16x4_f32(false, a, false, b,
                                                        (short)0, acc, false, false);
        }
        const float bb = bo[col];
#pragma unroll
        for (int v = 0; v < 8; ++v) {
            int row = v + 8 * hf;
            Xout[(long long)(r0 + row) * EMB + col] = lrelu(acc[v] + bb);
        }
    }
}

// ---------------------------------------------------------------------------
extern "C" void kernel_launch(void* const* d_in, const int* in_sizes, int n_in,
                              void* d_out, int out_size, void* d_ws, size_t ws_size,
                              hipStream_t stream) {
    const float* feat          = (const float*)d_in[0];
    const float* node_emb      = (const float*)d_in[1];
    const int*   src           = (const int*)  d_in[2];
    const int*   dst           = (const int*)  d_in[3];
    const float* user_feat_emb = (const float*)d_in[4];
    const float* trans_w       = (const float*)d_in[5];
    const float* trans_b       = (const float*)d_in[6];
    const float* ws            = (const float*)d_in[7];
    const float* des_w         = (const float*)d_in[8];
    const float* des_b         = (const float*)d_in[9];
    const float* out_w         = (const float*)d_in[10];
    const float* out_b         = (const float*)d_in[11];

    float* out   = (float*)d_out;
    float* outX  = out;                                   // [N_NODES, EMB]
    float* outUE = out + (size_t)N_NODES * EMB;           // [NUM_USERS, HID]

    // scratch: X (51.2 MB) + AGG (51.2 MB)
    float* X   = (float*)d_ws;
    float* AGG = X + (size_t)N_NODES * HID;

    // 1) users -> X[0:NUM_USERS) and second output
    gcn_copy_users<<<(NUM_USERS * HID / 4 + 255) / 256, 256, 0, stream>>>(
        user_feat_emb, X, outUE);

    // 2) items: X[NUM_USERS:) = feat @ trans_w + trans_b
    gcn_trans_items<<<NUM_ITEMS / 16, 256, 0, stream>>>(feat, trans_w, trans_b, X);

    // 3) row L2 normalize
    gcn_l2norm<<<N_NODES / 8, 256, 0, stream>>>(X);

    // 4) K layers
    for (int i = 0; i < NLAYERS; ++i) {
        gcn_zero<<<(N_NODES * HID / 4 + 255) / 256, 256, 0, stream>>>(
            (float4*)AGG, N_NODES * HID / 4);
        gcn_edge_scatter<<<(int)(((long long)N_EDGES * 32) / 256), 256, 0, stream>>>(
            src, dst, X, AGG);
        float* dstX = (i == NLAYERS - 1) ? outX : X;
        gcn_layer<<<N_NODES / 16, 256, 0, stream>>>(
            AGG,
            ws    + (size_t)i * HID * HID,
            des_w + (size_t)i * HID * EMB,
            des_b + (size_t)i * EMB,
            out_w + (size_t)i * (HID + EMB) * EMB,
            out_b + (size_t)i * EMB,
            node_emb, dstX);
    }
}